// ScaledDotProductAttention_16999480558402
// MI455X (gfx1250) — compile-verified
//
#include <hip/hip_runtime.h>

typedef float v2f __attribute__((ext_vector_type(2)));
typedef float v8f __attribute__((ext_vector_type(8)));

#define BATCH 2
#define HN    8
#define SN    256
#define DN    64
#define ROWS  2048            // H*S rows per batch
#define GR    256             // G rows per batch
#define GSZ   (GR * DN)       // 16384 floats per batch

// ---------------------------------------------------------------------------
// Kernel 1: G[b,r,k] = sum_{v=0..7} Q[b,8r+v,k] * V[b,8r+v,k]
// Stored in WMMA-friendly interleaved pair layout:
//   Gp[b][rp][k][j] = G[b, 2*rp + j, k]   (j in {0,1})
// so a WMMA-B operand {G[r][col], G[r+1][col]} is one contiguous 8B word.
// Each thread computes one (b, rp, k): 16 input rows, one float2 store.
// ---------------------------------------------------------------------------
__global__ void __launch_bounds__(256)
qv_group_sum_kernel(const float* __restrict__ Q,
                    const float* __restrict__ V,
                    float* __restrict__ Gp) {
    const int idx = blockIdx.x * 256 + threadIdx.x;   // [0, 2*128*64)
    const int b  = idx >> 13;                         // / 8192
    const int rp = (idx >> 6) & 127;                  // pair-row
    const int k  = idx & 63;
    const size_t base = ((size_t)b * ROWS + (size_t)rp * 16) * DN + k;
    float acc0 = 0.0f, acc1 = 0.0f;
#pragma unroll
    for (int v = 0; v < 8; ++v) {
        acc0 = fmaf(Q[base + (size_t)v * DN],       V[base + (size_t)v * DN],       acc0);
        acc1 = fmaf(Q[base + (size_t)(v + 8) * DN], V[base + (size_t)(v + 8) * DN], acc1);
    }
    float2 out;
    out.x = acc0;   // r = 2*rp
    out.y = acc1;   // r = 2*rp + 1
    *(float2*)(Gp + (size_t)b * GSZ + rp * (2 * DN) + k * 2) = out;
}

// ---------------------------------------------------------------------------
// Kernel 2: scores = M * G  (M is the 0/1 window-selection matrix, built in
// registers) via V_WMMA_F32_16X16X4_F32, then row softmax over d=64 and the
// fused elementwise outputs  attn  and  context = attn .* V.
// One wave computes a 16-row x 64-col tile: 4 accumulators, K-loop over 256.
// B operands are single ds_load_b64's thanks to the pair-interleaved layout.
// ---------------------------------------------------------------------------
__global__ void __launch_bounds__(256)
mask_matmul_softmax_kernel(const float* __restrict__ Gp,
                           const float* __restrict__ Vin,
                           float* __restrict__ out_ctx,
                           float* __restrict__ out_attn) {
    __shared__ float Gs[GSZ];                       // 64 KB of the 320 KB WGP LDS

    const int b        = blockIdx.x >> 4;           // grid = BATCH*16
    const int blockRow = blockIdx.x & 15;
    const int tid      = threadIdx.x;

    // Stage this batch's Gp (layout-preserving) into LDS, 128b transactions.
    {
        const float4* gsrc = (const float4*)(Gp + (size_t)b * GSZ);
        float4*       gdst = (float4*)Gs;
#pragma unroll
        for (int i = 0; i < GSZ / 4 / 256; ++i) {
            gdst[tid + i * 256] = gsrc[tid + i * 256];
        }
    }
    __syncthreads();

    const int wave    = tid >> 5;
    const int lane    = tid & 31;
    const int rowBase = blockRow * 128 + wave * 16; // 16 output rows per wave
    const int m       = lane & 15;                  // A-matrix row of this lane
    const int row     = rowBase + m;                // output row in [0,2048)
    const int h       = row >> 8;
    const int sIdx    = row & 255;
    // window base: r is selected iff ((r - wbase) mod 512) < 64
    const int wbase = (((sIdx & 7) << 6) + (h << 5) + (sIdx >> 3) + 256) & 511;
    const int kHalf = (lane < 16) ? 0 : 1;          // which K-pair this lane holds
    const int n     = lane & 15;                    // B/D column of this lane

    // LDS base for this lane: pair-row (k0/2 + kHalf), column n
    const float* gLane = Gs + (size_t)kHalf * (2 * DN) + n * 2;

    v8f acc0 = {}, acc1 = {}, acc2 = {}, acc3 = {};

#pragma unroll 4
    for (int k0 = 0; k0 < GR; k0 += 4) {
        const int r0 = k0 + kHalf * 2;
        // A tile: mask values (exact 0/1), ISA 16x4 f32 layout
        v2f a;
        a.x = (((r0     - wbase) & 511) < 64) ? 1.0f : 0.0f;
        a.y = (((r0 + 1 - wbase) & 511) < 64) ? 1.0f : 0.0f;
        // B tiles: {G[r0][col], G[r0+1][col]} is one aligned 8B LDS word
        const float* g = gLane + (k0 >> 1) * (2 * DN);
        v2f b0 = *(const v2f*)(g + 0);   // col n
        v2f b1 = *(const v2f*)(g + 32);  // col n+16
        v2f b2 = *(const v2f*)(g + 64);  // col n+32
        v2f b3 = *(const v2f*)(g + 96);  // col n+48
        acc0 = __builtin_amdgcn_wmma_f32_16x16x4_f32(false, a, false, b0, (short)0, acc0, false, false);
        acc1 = __builtin_amdgcn_wmma_f32_16x16x4_f32(false, a, false, b1, (short)0, acc1, false, false);
        acc2 = __builtin_amdgcn_wmma_f32_16x16x4_f32(false, a, false, b2, (short)0, acc2, false, false);
        acc3 = __builtin_amdgcn_wmma_f32_16x16x4_f32(false, a, false, b3, (short)0, acc3, false, false);
    }

    // Softmax over the 64 columns of each of this wave's 16 rows.
    // C/D layout: vgpr i <-> row (i or i+8), lanes 0-15 / 16-31 are the two
    // row halves; each row's 64 values live in 16 lanes x 4 accumulators, so
    // reduce across xor masks 1,2,4,8 (stays within each 16-lane half).
    const int rowHalf = rowBase + ((lane < 16) ? 0 : 8);
#pragma unroll
    for (int i = 0; i < 8; ++i) {
        const float s0 = acc0[i], s1 = acc1[i], s2 = acc2[i], s3 = acc3[i];
        float mx = fmaxf(fmaxf(s0, s1), fmaxf(s2, s3));
        mx = fmaxf(mx, __shfl_xor(mx, 1, 32));
        mx = fmaxf(mx, __shfl_xor(mx, 2, 32));
        mx = fmaxf(mx, __shfl_xor(mx, 4, 32));
        mx = fmaxf(mx, __shfl_xor(mx, 8, 32));
        const float e0 = __expf(s0 - mx);
        const float e1 = __expf(s1 - mx);
        const float e2 = __expf(s2 - mx);
        const float e3 = __expf(s3 - mx);
        float sum = (e0 + e1) + (e2 + e3);
        sum += __shfl_xor(sum, 1, 32);
        sum += __shfl_xor(sum, 2, 32);
        sum += __shfl_xor(sum, 4, 32);
        sum += __shfl_xor(sum, 8, 32);
        const float inv = 1.0f / sum;

        const int    ro = rowHalf + i;
        const size_t o  = ((size_t)b * ROWS + ro) * DN + n;
        const float a0 = e0 * inv, a1 = e1 * inv, a2 = e2 * inv, a3 = e3 * inv;
        out_attn[o]      = a0;  out_ctx[o]      = a0 * Vin[o];
        out_attn[o + 16] = a1;  out_ctx[o + 16] = a1 * Vin[o + 16];
        out_attn[o + 32] = a2;  out_ctx[o + 32] = a2 * Vin[o + 32];
        out_attn[o + 48] = a3;  out_ctx[o + 48] = a3 * Vin[o + 48];
    }
}

extern "C" void kernel_launch(void* const* d_in, const int* in_sizes, int n_in,
                              void* d_out, int out_size, void* d_ws, size_t ws_size,
                              hipStream_t stream) {
    (void)in_sizes; (void)n_in; (void)out_size; (void)ws_size;
    const float* Q = (const float*)d_in[0];
    const float* K = (const float*)d_in[1];   // unused: reference never reads K
    const float* V = (const float*)d_in[2];
    (void)K;

    float* Gp       = (float*)d_ws;                              // 128 KB scratch
    float* out_ctx  = (float*)d_out;                             // context first
    float* out_attn = (float*)d_out + (size_t)BATCH * ROWS * DN; // then attn

    // Kernel 1: 2*128*64 = 16384 threads (one per (b, pair-row, k))
    qv_group_sum_kernel<<<64, 256, 0, stream>>>(Q, V, Gp);
    // Kernel 2: BATCH * 16 blocks x 8 waves, each wave a 16x64 output tile
    mask_matmul_softmax_kernel<<<BATCH * 16, 256, 0, stream>>>(Gp, V, out_ctx, out_attn);
}